// GaussianOrbital_74990128988465
// MI455X (gfx1250) — compile-verified
//
#include <hip/hip_runtime.h>

// GaussianOrbital density on MI455X (gfx1250, wave32).
// r2 tile (16 grid-points x 16 atoms) computed with V_WMMA_F32_16X16X4_F32:
//   A[k,:] = (px,py,pz,1), B[:,n] = (-2xn,-2yn,-2zn,|xn|^2), C[k,n] = |pk|^2
//   => D = r2 exactly.
// Then per (k,n): density += w[g] * exp2(r2 * (-log2e/sigma_g^2)), w folded
// with 1/(sigma*sqrt(2pi))^3.  exp2 -> native v_exp_f32 (trans pipe),
// co-executes with VALU fmas; WMMA offloads r2 to the matrix pipe.

typedef __attribute__((ext_vector_type(2))) float v2f;
typedef __attribute__((ext_vector_type(8))) float v8f;

#define N_GAUSS   16
#define N_GRAPHS  64
#define K_GRID    4096

// sigma_g = linspace(0.5, 5.0, 16) -> 0.5 + 0.3*g  (all constant-folded)
__device__ constexpr float sigma_of(int g) { return 0.5f + 0.3f * (float)g; }
// weight scale: 1/(sigma*sqrt(2pi))^3 = 1/(sigma^3 * (2pi)^1.5)
__device__ constexpr float srg_of(int g) {
  float s = sigma_of(g);
  return 1.0f / (s * s * s * 15.749609945722419f);
}
// exp2 scale: -log2(e)/sigma^2
__device__ constexpr float e2_of(int g) {
  float s = sigma_of(g);
  return -1.4426950408889634f / (s * s);
}

__global__ __launch_bounds__(256)
void gauss_density_kernel(const float* __restrict__ coeff,       // (N,16)
                          const float* __restrict__ atom_coord,  // (N,3)
                          const float* __restrict__ grid,        // (64,4096,3)
                          const int*  __restrict__ batch,        // (N,) sorted
                          float* __restrict__ out,               // (64,4096)
                          int n_atoms)
{
  const int g    = blockIdx.x;            // graph id
  const int lane = threadIdx.x & 31;
  const int wave = threadIdx.x >> 5;
  const int kbase = blockIdx.y * 128 + wave * 16;   // this wave's 16-k tile
  const int kk = lane & 15;

  // --- segment [astart, aend) of atoms in graph g (batch sorted) ---
  int lo = 0, hi = n_atoms;
  while (lo < hi) { int m = (lo + hi) >> 1; if (batch[m] <  g) lo = m + 1; else hi = m; }
  const int astart = lo;
  hi = n_atoms;
  while (lo < hi) { int m = (lo + hi) >> 1; if (batch[m] <= g) lo = m + 1; else hi = m; }
  const int aend = lo;

  // --- per-k-tile setup: A operand and C = |p|^2 tile ---
  const float* gp = grid + ((size_t)g * K_GRID + kbase + kk) * 3;
  const float px = gp[0], py = gp[1], pz = gp[2];
  const float pp = px * px + py * py + pz * pz;

  // A 16x4 f32 layout: lanes0-15 VGPR{0,1}=K{0,1}; lanes16-31 VGPR{0,1}=K{2,3}
  v2f A;
  if (lane < 16) { A.x = px; A.y = py; }
  else           { A.x = pz; A.y = 1.0f; }

  // C/D layout: VGPR r holds row M=r (lanes0-15) / M=r+8 (lanes16-31), N=lane%16
  v8f Cpp;
#pragma unroll
  for (int r = 0; r < 8; ++r)
    Cpp[r] = __shfl(pp, (lane & 16) ? (8 + r) : r, 32);

  float acc[8];
#pragma unroll
  for (int r = 0; r < 8; ++r) acc[r] = 0.0f;

  // --- loop over atom tiles of 16 ---
  for (int atile = astart; atile < aend; atile += 16) {
    const int a = atile + kk;
    const bool valid = a < aend;
    float ax = 0.0f, ay = 0.0f, az = 0.0f;   // pad: coord 0 -> r2 finite
    if (valid) {
      const float* ap = atom_coord + 3 * (size_t)a;
      ax = ap[0]; ay = ap[1]; az = ap[2];
    }
    // B 4x16 f32 layout mirrors A: lanes0-15 hold rows K=0,1; lanes16-31 K=2,3
    v2f B;
    if (lane < 16) { B.x = -2.0f * ax; B.y = -2.0f * ay; }
    else           { B.x = -2.0f * az; B.y = ax * ax + ay * ay + az * az; }

    // per-lane weights for its atom column; padded atoms get w = 0
    float w[16];
    const float* cw = coeff + (size_t)a * N_GAUSS;
#pragma unroll
    for (int q = 0; q < 16; q += 4) {
      float4 c4 = valid ? *(const float4*)(cw + q) : make_float4(0.f, 0.f, 0.f, 0.f);
      w[q + 0] = c4.x * srg_of(q + 0);
      w[q + 1] = c4.y * srg_of(q + 1);
      w[q + 2] = c4.z * srg_of(q + 2);
      w[q + 3] = c4.w * srg_of(q + 3);
    }

    // r2 tile: one WMMA (EXEC is full here; no divergent flow encloses it)
    v8f r2 = __builtin_amdgcn_wmma_f32_16x16x4_f32(
        false, A, false, B, (short)0, Cpp, false, false);

#pragma unroll
    for (int r = 0; r < 8; ++r) {
      float d = acc[r];
#pragma unroll
      for (int gg = 0; gg < N_GAUSS; ++gg)
        d += w[gg] * __builtin_amdgcn_exp2f(r2[r] * e2_of(gg));
      acc[r] = d;
    }
  }

  // --- reduce across the 16 atom columns in each half-wave ---
#pragma unroll
  for (int off = 1; off < 16; off <<= 1) {
#pragma unroll
    for (int r = 0; r < 8; ++r)
      acc[r] += __shfl_xor(acc[r], off, 32);
  }

  // lane 0 writes k rows 0..7, lane 16 writes rows 8..15
  if ((lane & 15) == 0) {
    float* o = out + (size_t)g * K_GRID + kbase + ((lane >> 4) << 3);
#pragma unroll
    for (int r = 0; r < 8; ++r) o[r] = acc[r];
  }
}

extern "C" void kernel_launch(void* const* d_in, const int* in_sizes, int n_in,
                              void* d_out, int out_size, void* d_ws, size_t ws_size,
                              hipStream_t stream) {
  const float* coeff      = (const float*)d_in[0];
  const float* atom_coord = (const float*)d_in[1];
  const float* grid       = (const float*)d_in[2];
  const int*   batch      = (const int*)d_in[3];
  float* out = (float*)d_out;
  const int n_atoms = in_sizes[3];

  dim3 blocks(N_GRAPHS, K_GRID / 128);   // 64 graphs x 32 k-chunks of 128
  gauss_density_kernel<<<blocks, dim3(256), 0, stream>>>(
      coeff, atom_coord, grid, batch, out, n_atoms);
}